// SWD18_28449863369562
// MI455X (gfx1250) — compile-verified
//
#include <hip/hip_runtime.h>

// Sorted-window-dispersion kernel for MI455X (gfx1250).
// out[b,t,i] = sort-within-runs-of-5(v[b,:,i]) where runs start at t ≡ i (mod 5).
// (The per-feature cyclic shift/unshift in the reference cancel in addressing.)
//
// Memory-bound: 131 MB total traffic -> ~5.6 us roofline at 23.3 TB/s.
// CDNA5 path: GLOBAL_LOAD_ASYNC_TO_LDS_B128 / GLOBAL_STORE_ASYNC_FROM_LDS_B128
// (ASYNCcnt) for the bulk coalesced traffic.

#define WIN       5
#define TILE_R    40                 // window-start rows per tile (multiple of WIN)
#define HALO      4                  // max phase offset
#define TILE_ROWS (TILE_R + HALO)    // 44 rows resident in LDS
#define TILE_C    256                // feature columns per block (= blockDim.x)

// ---- CDNA5 async global<->LDS movers (ASYNCcnt-tracked), with safe fallback ----
#if defined(__has_builtin)
#if __has_builtin(__builtin_amdgcn_global_load_async_to_lds_b128) && \
    __has_builtin(__builtin_amdgcn_global_store_async_from_lds_b128)
#define USE_ASYNC_LDS 1
#endif
#if __has_builtin(__builtin_amdgcn_s_wait_asynccnt)
#define HAVE_WAIT_ASYNC_BUILTIN 1
#endif
#endif

#if defined(USE_ASYNC_LDS)
typedef int v4i __attribute__((ext_vector_type(4)));
typedef __attribute__((address_space(1))) v4i GV4;   // global-AS int4
typedef __attribute__((address_space(3))) v4i LV4;   // LDS-AS int4
#endif

__device__ __forceinline__ void wait_async0() {
#if defined(HAVE_WAIT_ASYNC_BUILTIN)
    __builtin_amdgcn_s_wait_asynccnt(0);
#else
    asm volatile("s_wait_asynccnt 0x0" ::: "memory");
#endif
}

#define CSWAP(x, y)                          \
    do {                                     \
        float _lo = fminf((x), (y));         \
        float _hi = fmaxf((x), (y));         \
        (x) = _lo;                           \
        (y) = _hi;                           \
    } while (0)

__global__ __launch_bounds__(TILE_C) void swd_window_sort_kernel(
    const float* __restrict__ v, float* __restrict__ out, int L, int D)
{
    __shared__ __align__(16) float tile[TILE_ROWS * TILE_C];

    const int tid = threadIdx.x;
    const int i0  = blockIdx.x * TILE_C;     // first feature column of this block
    const int t0  = blockIdx.y * TILE_R;     // first window-start row (t0 % 5 == 0)
    const int b   = blockIdx.z;

    const size_t plane = (size_t)L * (size_t)D;
    const float* vb = v   + (size_t)b * plane;
    float*       ob = out + (size_t)b * plane;

    // ---------- Phase 1: async-load tile rows [t0, t0+44) (mod L), 1 KB per row ----------
    const int chunksPerRow = TILE_C / 4;                     // 64 x b128 per row
    for (int idx = tid; idx < TILE_ROWS * chunksPerRow; idx += TILE_C) {
        const int slot = idx / chunksPerRow;
        const int c4   = (idx % chunksPerRow) * 4;
        int row = t0 + slot;
        if (row >= L) row -= L;                              // circular halo wrap
        const float* gsrc = vb + (size_t)row * D + (size_t)(i0 + c4);
        float*       lptr = &tile[slot * TILE_C + c4];
#if defined(USE_ASYNC_LDS)
        __builtin_amdgcn_global_load_async_to_lds_b128(
            (GV4*)const_cast<float*>(gsrc), (LV4*)lptr, 0, 0);
#else
        *(float4*)lptr = *(const float4*)gsrc;
#endif
    }
#if defined(USE_ASYNC_LDS)
    wait_async0();
#endif
    __syncthreads();

    // ---------- Phase 2: per-column 5-element sorting networks in LDS ----------
    const int i     = i0 + tid;
    const int phase = i % WIN;                               // run starts at t ≡ i (mod 5)
#pragma unroll
    for (int k = 0; k < TILE_R / WIN; ++k) {
        float* p = &tile[(phase + WIN * k) * TILE_C + tid];  // bank = tid%64: conflict-free
        float a0 = p[0 * TILE_C];
        float a1 = p[1 * TILE_C];
        float a2 = p[2 * TILE_C];
        float a3 = p[3 * TILE_C];
        float a4 = p[4 * TILE_C];
        // optimal 9-comparator network for n=5
        CSWAP(a0, a1); CSWAP(a3, a4); CSWAP(a2, a4);
        CSWAP(a2, a3); CSWAP(a1, a4); CSWAP(a0, a3);
        CSWAP(a0, a2); CSWAP(a1, a3); CSWAP(a1, a2);
        p[0 * TILE_C] = a0;
        p[1 * TILE_C] = a1;
        p[2 * TILE_C] = a2;
        p[3 * TILE_C] = a3;
        p[4 * TILE_C] = a4;
    }
    __syncthreads();

    // ---------- Phase 3a: bulk async-store interior rows (final for every phase) ----------
    for (int idx = tid; idx < (TILE_R - HALO) * chunksPerRow; idx += TILE_C) {
        const int slot = HALO + idx / chunksPerRow;          // 4..39, never wraps
        const int c4   = (idx % chunksPerRow) * 4;
        const int row  = t0 + slot;
        float* gdst = ob + (size_t)row * D + (size_t)(i0 + c4);
        float* lptr = &tile[slot * TILE_C + c4];
#if defined(USE_ASYNC_LDS)
        __builtin_amdgcn_global_store_async_from_lds_b128(
            (GV4*)gdst, (LV4*)lptr, 0, 0);
#else
        *(float4*)gdst = *(const float4*)lptr;
#endif
    }

    // ---------- Phase 3b: edge rows, predicated per-element (each written once grid-wide) ----------
#pragma unroll
    for (int s = 0; s < HALO; ++s) {
        if (phase <= s) {                                    // low edge: slot s owned iff phase <= s
            const int row = t0 + s;                          // t0+3 < L, no wrap
            ob[(size_t)row * D + i] = tile[s * TILE_C + tid];
        }
        if (s < phase) {                                     // high edge: slot TILE_R+s owned iff s < phase
            int row = t0 + TILE_R + s;
            if (row >= L) row -= L;
            ob[(size_t)row * D + i] = tile[(TILE_R + s) * TILE_C + tid];
        }
    }

#if defined(USE_ASYNC_LDS)
    wait_async0();   // drain async stores (s_endpgm would also wait-idle)
#endif
}

extern "C" void kernel_launch(void* const* d_in, const int* in_sizes, int n_in,
                              void* d_out, int out_size, void* d_ws, size_t ws_size,
                              hipStream_t stream) {
    // Inputs: q (unused), k (unused), v — all float32 [B, L, D].
    const float* v  = (const float*)d_in[2];
    float*      out = (float*)d_out;

    const int L = 4000;
    const int D = 512;
    const int B = (int)((long long)out_size / ((long long)L * D));   // = 8

    dim3 block(TILE_C);
    dim3 grid(D / TILE_C, L / TILE_R, B);                            // 2 x 100 x 8
    swd_window_sort_kernel<<<grid, block, 0, stream>>>(v, out, L, D);
}